// SparseMaskController_57226144252249
// MI455X (gfx1250) — compile-verified
//
#include <hip/hip_runtime.h>
#include <hip/hip_bf16.h>

// Sizes
#define B_      8
#define S_      4096
#define HID_    2048
#define ADIM_   32
#define TRANK_  16
#define SCHUNKS 32          // 4096/32 = 128 rows per partial
#define HCH4    512         // 2048 floats = 512 float4 per row

typedef __attribute__((ext_vector_type(2))) float v2f;
typedef __attribute__((ext_vector_type(8))) float v8f;

// ---------------------------------------------------------------------------
// Phase A1: partial column sums of hidden_states over S (deterministic, no atomics)
// grid (SCHUNKS, 2, B), block 256.  Each thread sums 128 rows of one float4 column.
// ---------------------------------------------------------------------------
__global__ void pool_partial_kernel(const float* __restrict__ hs,
                                    float* __restrict__ partials) {
    const int tid = threadIdx.x;
    const int sc  = blockIdx.x;      // s-chunk
    const int hc  = blockIdx.y;      // h-chunk (of float4s)
    const int b   = blockIdx.z;
    const int h4  = hc * 256 + tid;  // [0, 512)

    const float4* base = (const float4*)hs + ((size_t)b * S_ + (size_t)sc * 128) * HCH4 + h4;
    float4 acc = make_float4(0.f, 0.f, 0.f, 0.f);
    for (int s = 0; s < 128; ++s) {
        float4 v = base[(size_t)s * HCH4];
        acc.x += v.x; acc.y += v.y; acc.z += v.z; acc.w += v.w;
    }
    ((float4*)partials)[((size_t)(b * SCHUNKS + sc)) * HCH4 + h4] = acc;
}

// ---------------------------------------------------------------------------
// Phase A2: reduce partials -> pooled mean.  grid (8, B), block 256.
// ---------------------------------------------------------------------------
__global__ void pool_reduce_kernel(const float* __restrict__ partials,
                                   float* __restrict__ pooled) {
    const int tid = threadIdx.x;
    const int hc  = blockIdx.x;      // [0,8)
    const int b   = blockIdx.y;
    const int h   = hc * 256 + tid;  // [0,2048)

    float acc = 0.f;
    for (int c = 0; c < SCHUNKS; ++c)
        acc += partials[((size_t)(b * SCHUNKS + c)) * HID_ + h];
    pooled[b * HID_ + h] = acc * (1.0f / (float)S_);
}

// ---------------------------------------------------------------------------
// Phase B: controller MLP + LN + exact GELU + top-8 mask.
// Single block of 256 threads; wave w handles batch w (wave32), lane = feature j.
// Output: wbuf[b*16 + r] = mask * rank_scales[r] * (1/32)   (Hadamard magnitude folded in)
// ---------------------------------------------------------------------------
__global__ void controller_kernel(const float* __restrict__ pooled,
                                  const float* __restrict__ W1,
                                  const float* __restrict__ b1,
                                  const float* __restrict__ ln_gamma,
                                  const float* __restrict__ ln_beta,
                                  const float* __restrict__ W2,
                                  const float* __restrict__ b2,
                                  const float* __restrict__ mask_logits,
                                  const float* __restrict__ rank_scales,
                                  float* __restrict__ wbuf) {
    __shared__ float sx[B_ * ADIM_];
    __shared__ float scomb[B_ * TRANK_];

    const int tid = threadIdx.x;
    const int b = tid >> 5;          // batch per wave
    const int j = tid & 31;          // feature

    // x = pooled[b] @ W1 + b1   (uniform pooled reads broadcast within the wave)
    float acc = b1[j];
    const float* pb = pooled + b * HID_;
    for (int k = 0; k < HID_; ++k)
        acc += pb[k] * W1[k * ADIM_ + j];

    sx[tid] = acc;
    __syncthreads();

    // LayerNorm over ADIM_ (population variance, as jnp.var)
    float s1 = 0.f, s2 = 0.f;
    for (int q = 0; q < ADIM_; ++q) {
        float v = sx[b * ADIM_ + q];
        s1 += v; s2 += v * v;
    }
    const float mu  = s1 * (1.0f / ADIM_);
    const float var = s2 * (1.0f / ADIM_) - mu * mu;
    float xn = (acc - mu) * rsqrtf(var + 1e-5f) * ln_gamma[j] + ln_beta[j];

    // exact GELU
    float g = 0.5f * xn * (1.0f + erff(xn * 0.70710678118654752440f));
    __syncthreads();
    sx[tid] = g;
    __syncthreads();

    // logits = g @ W2 + b2 + mask_logits
    if (j < TRANK_) {
        float lg = b2[j];
        for (int q = 0; q < ADIM_; ++q)
            lg += sx[b * ADIM_ + q] * W2[q * TRANK_ + j];
        scomb[b * TRANK_ + j] = lg + mask_logits[j];
    }
    __syncthreads();

    // top-8 mask (jax.lax.top_k tie-break: lower index wins)
    if (j < TRANK_) {
        const float c = scomb[b * TRANK_ + j];
        int cnt = 0;
        for (int q = 0; q < TRANK_; ++q) {
            float cq = scomb[b * TRANK_ + q];
            cnt += ((cq > c) || (cq == c && q < j)) ? 1 : 0;
        }
        // fold Hadamard magnitude 1/32 in here
        wbuf[b * TRANK_ + j] = (cnt < 8) ? rank_scales[j] * 0.03125f : 0.0f;
    }
}

// ---------------------------------------------------------------------------
// Phase C: final[t,:] = (act[t,:] * w[b,:]) @ H via V_WMMA_F32_16X16X4_F32.
// One wave per 16-token tile; four K=4 steps accumulate the 16x16 product.
// B' rows carry w[b][k] * sign(k,n) * (1/32)  (magnitude already in wbuf).
//   A layout : lane (hi,lo): row M=lo, K = 4s + 2*hi + {0,1}  (v2f)
//   B layout : lane (hi,lo): col N=lo, K = 4s + 2*hi + {0,1}  (v2f)
//   D layout : VGPR j: lanes 0-15 row j, lanes 16-31 row j+8, N=lo (v8f)
// ---------------------------------------------------------------------------
__global__ void hadamard_apply_kernel(const float* __restrict__ act,
                                      const float* __restrict__ wbuf,
                                      float* __restrict__ out) {
    const int lane = threadIdx.x & 31;
    const int wid  = (blockIdx.x * blockDim.x + threadIdx.x) >> 5;
    const int nw   = (gridDim.x * blockDim.x) >> 5;
    const int hi   = lane >> 4;      // 0 or 1
    const int lo   = lane & 15;

    const int ntiles = (B_ * S_) / 16;   // 2048

    for (int tile = wid; tile < ntiles; tile += nw) {
        const int token0 = tile << 4;            // global token index
        const int b = token0 >> 12;              // 4096 tokens per batch
        const float* wb   = wbuf + b * TRANK_;
        const float* arow = act + (size_t)(token0 + lo) * TRANK_;

        v8f c = {0.f, 0.f, 0.f, 0.f, 0.f, 0.f, 0.f, 0.f};

#pragma unroll
        for (int s = 0; s < 4; ++s) {
            const int k0 = s * 4 + hi * 2;
            // A fragment: two consecutive-K elements of row `lo`
            v2f a = *(const v2f*)(arow + k0);
            // B' fragment: w[k] with Hadamard sign (-1)^popcount(k & n)
            float w0 = wb[k0];
            float w1 = wb[k0 + 1];
            int sg0 = (__popc(k0 & lo) & 1) << 31;
            int sg1 = (__popc((k0 + 1) & lo) & 1) << 31;
            v2f bb;
            bb.x = __int_as_float(__float_as_int(w0) ^ sg0);
            bb.y = __int_as_float(__float_as_int(w1) ^ sg1);

            c = __builtin_amdgcn_wmma_f32_16x16x4_f32(
                    /*neg_a=*/false, a, /*neg_b=*/false, bb,
                    /*c_mod=*/(short)0, c, /*reuse_a=*/false, /*reuse_b=*/false);
        }

        float* orow = out + (size_t)token0 * TRANK_;
#pragma unroll
        for (int jj = 0; jj < 8; ++jj) {
            orow[(jj + hi * 8) * TRANK_ + lo] = c[jj];   // contiguous 64B per half-wave
        }
    }
}

// ---------------------------------------------------------------------------
extern "C" void kernel_launch(void* const* d_in, const int* in_sizes, int n_in,
                              void* d_out, int out_size, void* d_ws, size_t ws_size,
                              hipStream_t stream) {
    const float* hidden_states    = (const float*)d_in[0];
    const float* rank_activations = (const float*)d_in[1];
    const float* W1               = (const float*)d_in[2];
    const float* b1               = (const float*)d_in[3];
    const float* ln_gamma         = (const float*)d_in[4];
    const float* ln_beta          = (const float*)d_in[5];
    const float* W2               = (const float*)d_in[6];
    const float* b2               = (const float*)d_in[7];
    const float* mask_logits      = (const float*)d_in[8];
    const float* rank_scales      = (const float*)d_in[9];
    float* out = (float*)d_out;

    // workspace layout
    float* partials = (float*)d_ws;                       // B_*SCHUNKS*HID_ = 524288 f
    float* pooled   = partials + (size_t)B_ * SCHUNKS * HID_; // 16384 f
    float* wbuf     = pooled + (size_t)B_ * HID_;             // 128 f

    // A1: partial pooling (dominant 256 MiB read)
    pool_partial_kernel<<<dim3(SCHUNKS, 2, B_), 256, 0, stream>>>(hidden_states, partials);
    // A2: finish mean
    pool_reduce_kernel<<<dim3(8, B_), 256, 0, stream>>>(partials, pooled);
    // B: controller -> per-batch rank weights
    controller_kernel<<<1, 256, 0, stream>>>(pooled, W1, b1, ln_gamma, ln_beta,
                                             W2, b2, mask_logits, rank_scales, wbuf);
    // C: masked/scaled Hadamard transform via WMMA
    hadamard_apply_kernel<<<128, 256, 0, stream>>>(rank_activations, wbuf, out);
}